// SegHead_20109036880092
// MI455X (gfx1250) — compile-verified
//
#include <hip/hip_runtime.h>
#include <stdint.h>

// Problem constants (from the reference)
#define BATCH  2
#define DDIM   32
#define NELEM  (256 * 256 * 16)   // 1048576 per batch row
#define KSEL   500
#define TILE_N 256                 // TDM tile width (columns per block)
#define HBINS  2048
#define ELEMS  4                   // elements per thread in hist/final kernels

typedef unsigned int u32x4 __attribute__((ext_vector_type(4)));
typedef int          i32x8 __attribute__((ext_vector_type(8)));
typedef int          i32x4 __attribute__((ext_vector_type(4)));

#if defined(__HIP_DEVICE_COMPILE__) && __has_builtin(__builtin_amdgcn_tensor_load_to_lds)
#define USE_TDM 1
#else
#define USE_TDM 0
#endif

__device__ __forceinline__ unsigned f2key(float f) {
  unsigned u = __float_as_uint(f);
  // monotonic unsigned ordering: larger float -> larger key
  return (u & 0x80000000u) ? ~u : (u | 0x80000000u);
}

__device__ __forceinline__ void wait_tensorcnt0() {
#if defined(__HIP_DEVICE_COMPILE__)
#if __has_builtin(__builtin_amdgcn_s_wait_tensorcnt)
  __builtin_amdgcn_s_wait_tensorcnt(0);
#else
  asm volatile("s_wait_tensorcnt 0" ::: "memory");
#endif
#endif
}

// ---------------------------------------------------------------------------
// Kernel 1: mean over D. grid = BATCH*NELEM/TILE_N blocks, TILE_N threads.
// Each block produces TILE_N consecutive outputs of one batch row.
// TDM path: one tensor_load_to_lds DMAs a [DDIM x TILE_N] f32 tile (32 KB)
// into LDS; each lane then reduces one column (bank-conflict-free).
// ---------------------------------------------------------------------------
__global__ void mean_kernel(const float* __restrict__ x, float* __restrict__ out) {
  const int tilesPerBatch = NELEM / TILE_N;
  const int b  = blockIdx.x / tilesPerBatch;
  const int n0 = (blockIdx.x % tilesPerBatch) * TILE_N;

#if USE_TDM
  __shared__ float tile[DDIM][TILE_N];   // 32 KB

  if (threadIdx.x < 32) {  // wave 0 issues the DMA (EXEC ignored by TDM)
    unsigned long long gaddr =
        (unsigned long long)(uintptr_t)x +
        ((unsigned long long)b * DDIM * NELEM + (unsigned long long)n0) * 4ull;
    unsigned ldsAddr = (unsigned)(uintptr_t)(&tile[0][0]);  // LDS byte offset (low 32 bits)

    // D# group 0: count=1, lds_addr, global_addr[56:0], type=2
    u32x4 g0;
    g0.x = 1u;
    g0.y = ldsAddr;
    g0.z = (unsigned)(gaddr & 0xFFFFFFFFull);
    g0.w = ((unsigned)((gaddr >> 32) & 0x01FFFFFFull)) | (2u << 30);

    // D# group 1: data_size=4B; tensor_dim0=TILE_N; tensor_dim1=DDIM;
    // tile_dim0=TILE_N; tile_dim1=DDIM; tensor_dim0_stride=NELEM elements.
    i32x8 g1;
    g1[0] = 0x00020000;                               // data_size = 2 (4 bytes)
    g1[1] = (int)(((unsigned)TILE_N & 0xFFFFu) << 16);// tensor_dim0[15:0]
    g1[2] = (int)(((unsigned)DDIM & 0xFFFFu) << 16);  // tensor_dim0[31:16]=0 | tensor_dim1[15:0]
    g1[3] = (int)(((unsigned)TILE_N) << 16);          // tensor_dim1[31:16]=0 | tile_dim0
    g1[4] = DDIM;                                     // tile_dim1 | tile_dim2=0
    g1[5] = NELEM;                                    // tensor_dim0_stride[31:0]
    g1[6] = 0;                                        // stride0[47:32] | stride1[15:0]
    g1[7] = 0;                                        // stride1[47:16]

    i32x4 g2 = {0, 0, 0, 0};
    i32x4 g3 = {0, 0, 0, 0};
#if __clang_major__ >= 23
    i32x8 g4 = {0, 0, 0, 0, 0, 0, 0, 0};
    __builtin_amdgcn_tensor_load_to_lds(g0, g1, g2, g3, g4, 0);
#else
    __builtin_amdgcn_tensor_load_to_lds(g0, g1, g2, g3, 0);
#endif
    wait_tensorcnt0();
  }
  __syncthreads();
  // Tell the compiler LDS was written behind its back (by the TDM).
  asm volatile("" : : "r"(&tile[0][0]) : "memory");

  const int c = threadIdx.x;
  float s = 0.0f;
#pragma unroll
  for (int d = 0; d < DDIM; ++d) s += tile[d][c];
  out[(long long)b * NELEM + n0 + c] = s * (1.0f / (float)DDIM);

#else  // fallback: direct strided loads + prefetch of the next D-plane
  const long long base = (long long)b * DDIM * NELEM + n0 + threadIdx.x;
  float s = 0.0f;
#pragma unroll 4
  for (int d = 0; d < DDIM; ++d) {
    if (d + 1 < DDIM) __builtin_prefetch(&x[base + (long long)(d + 1) * NELEM], 0, 0);
    s += x[base + (long long)d * NELEM];
  }
  out[(long long)b * NELEM + n0 + threadIdx.x] = s * (1.0f / (float)DDIM);
#endif
}

// ---------------------------------------------------------------------------
// Kernel 2: radix-select histogram pass (MSB-first). LDS histogram with
// nonzero-only flush (Gaussian data makes a few bins extremely hot).
// grid = BATCH*NELEM/(256*ELEMS) blocks, 256 threads.
// ---------------------------------------------------------------------------
__global__ void hist_kernel(const float* __restrict__ mean, const unsigned* __restrict__ ctrl,
                            unsigned* __restrict__ hist, int pass, int shift, int width) {
  __shared__ unsigned lh[HBINS];
  for (int i = threadIdx.x; i < HBINS; i += blockDim.x) lh[i] = 0;
  __syncthreads();

  const int bpb = gridDim.x / BATCH;
  const int b = blockIdx.x / bpb;
  const long long base =
      (long long)b * NELEM + (long long)(blockIdx.x % bpb) * (blockDim.x * ELEMS);

  const unsigned prefix    = ctrl[b];
  const unsigned maskAbove = (pass == 0) ? 0u : ~((1u << (shift + width)) - 1u);
  const unsigned dmask     = (1u << width) - 1u;

#pragma unroll
  for (int k = 0; k < ELEMS; ++k) {
    const long long i = base + threadIdx.x + (long long)k * blockDim.x;
    const unsigned key = f2key(mean[i]);
    if ((key & maskAbove) == (prefix & maskAbove))
      atomicAdd(&lh[(key >> shift) & dmask], 1u);
  }
  __syncthreads();

  unsigned* g = hist + (size_t)(pass * BATCH + b) * HBINS;
  for (int i = threadIdx.x; i < HBINS; i += blockDim.x) {
    const unsigned c = lh[i];
    if (c) atomicAdd(&g[i], c);
  }
}

// ---------------------------------------------------------------------------
// Kernel 3: walk the histogram top-down to pick the digit holding the K-th
// largest; update prefix and remaining count. grid = BATCH blocks.
// ---------------------------------------------------------------------------
__global__ void scan_kernel(unsigned* __restrict__ ctrl, const unsigned* __restrict__ hist,
                            int pass, int shift, int width) {
  if (threadIdx.x != 0) return;
  const int b = blockIdx.x;
  const unsigned* h = hist + (size_t)(pass * BATCH + b) * HBINS;

  unsigned rem = (pass == 0) ? (unsigned)KSEL : ctrl[2 + b];
  const int nb = 1 << width;
  unsigned cum = 0, digit = 0;
  for (int bin = nb - 1; bin >= 0; --bin) {
    const unsigned c = h[bin];
    if (cum + c >= rem) { digit = (unsigned)bin; break; }
    cum += c;
  }
  ctrl[b]     |= digit << shift;   // accumulate threshold key bits
  ctrl[2 + b]  = rem - cum;        // still needed at/below this digit
}

// ---------------------------------------------------------------------------
// Kernel 4: in-place rewrite of d_out: key > T -> 10v; key == T -> keep the
// first `need` ties (atomic slot grab); else 0.
// ---------------------------------------------------------------------------
__global__ void final_kernel(float* __restrict__ out, unsigned* __restrict__ ctrl) {
  const int bpb = gridDim.x / BATCH;
  const int b = blockIdx.x / bpb;
  const long long base =
      (long long)b * NELEM + (long long)(blockIdx.x % bpb) * (blockDim.x * ELEMS);

  const unsigned T    = ctrl[b];
  const unsigned need = ctrl[2 + b];

#pragma unroll
  for (int k = 0; k < ELEMS; ++k) {
    const long long i = base + threadIdx.x + (long long)k * blockDim.x;
    const float v = out[i];
    const unsigned key = f2key(v);
    float r = 0.0f;
    if (key > T) {
      r = 10.0f * v;
    } else if (key == T) {
      const unsigned slot = atomicAdd(&ctrl[4 + b], 1u);
      if (slot < need) r = 10.0f * v;
    }
    out[i] = r;
  }
}

// ---------------------------------------------------------------------------
extern "C" void kernel_launch(void* const* d_in, const int* in_sizes, int n_in,
                              void* d_out, int out_size, void* d_ws, size_t ws_size,
                              hipStream_t stream) {
  (void)in_sizes; (void)n_in; (void)out_size; (void)ws_size;
  const float* x = (const float*)d_in[0];
  float* out = (float*)d_out;

  unsigned* ctrl = (unsigned*)d_ws;   // [0..1]=prefix/T, [2..3]=remaining/need, [4..5]=tie ctr
  unsigned* hist = ctrl + 16;         // 3 passes x BATCH x HBINS

  hipMemsetAsync(d_ws, 0, (16 + 3 * BATCH * HBINS) * sizeof(unsigned), stream);

  // Pass 1: means (staged in d_out), 256 MB streamed via TDM tiles.
  mean_kernel<<<BATCH * NELEM / TILE_N, TILE_N, 0, stream>>>(x, out);

  // Exact top-K threshold via 3-pass MSB radix select over float-order keys.
  const int shifts[3] = {21, 10, 0};
  const int widths[3] = {11, 11, 10};
  const int selGrid = BATCH * NELEM / (256 * ELEMS);
  for (int p = 0; p < 3; ++p) {
    hist_kernel<<<selGrid, 256, 0, stream>>>(out, ctrl, hist, p, shifts[p], widths[p]);
    scan_kernel<<<BATCH, 32, 0, stream>>>(ctrl, hist, p, shifts[p], widths[p]);
  }

  // Final scatter/zero rewrite in place.
  final_kernel<<<selGrid, 256, 0, stream>>>(out, ctrl);
}